// IRB_10118942949617
// MI455X (gfx1250) — compile-verified
//
#include <hip/hip_runtime.h>

typedef __attribute__((ext_vector_type(2))) float v2f;
typedef __attribute__((ext_vector_type(8))) float v8f;

#define NGRP   9
#define IG1    128
#define OG1    512
#define IG2    512
#define OG2    128
#define C_IN   1152
#define TM     16            // tokens per workgroup tile

#define XG_LD  132           // padded row stride (floats) for Xg -> bank-conflict-free
#define H_LD   4612          // padded row stride (floats) for H  -> bank-conflict-free
#define SMEM_FLOATS (TM * XG_LD + TM * H_LD)

__device__ __forceinline__ float hardswish_f(float v) {
    float t = fminf(fmaxf(v + 3.0f, 0.0f), 6.0f);
    return v * t * (1.0f / 6.0f);
}

__global__ __launch_bounds__(256, 1)
void fused_local_mlp(const float* __restrict__ x,
                     const float* __restrict__ W1,
                     const float* __restrict__ b1,
                     const float* __restrict__ W2,
                     const float* __restrict__ b2,
                     float* __restrict__ out) {
    extern __shared__ float smem[];
    float* Xg = smem;                 // [TM][XG_LD]
    float* H  = smem + TM * XG_LD;    // [TM][H_LD]

    const int tid  = threadIdx.x;
    const int wave = tid >> 5;
    const int lane = tid & 31;
    const int nn   = lane & 15;       // A-row index (m) == B/C column index (n)
    const int hh   = lane >> 4;       // lane half
    const int koff = hh * 2;          // K-pair offset within a 16x16x4 step

    const long token_base = (long)blockIdx.x * TM;

    // ================= Layer 1: X[16,1152] x W1 -> H[16,4608] (interleaved) ==
    // Each wave: 4 register-blocked output tiles per group (A reused 4x).
    for (int g = 0; g < NGRP; ++g) {
        __syncthreads();  // previous Xg fully consumed before overwrite

        // Stage Xg[16][128]: 512 float4s, coalesced
        #pragma unroll
        for (int j = 0; j < 2; ++j) {
            int idx4 = tid + j * 256;            // 0..511
            int m    = idx4 >> 5;                // 32 float4 per row
            int k4   = idx4 & 31;
            const float4 v = *(const float4*)(x + (token_base + m) * C_IN + g * IG1 + k4 * 4);
            *(float4*)(Xg + m * XG_LD + k4 * 4) = v;
        }
        __syncthreads();

        const int obase0 = wave * 64;            // 4 tiles: obase0 + u*16
        v8f c0 = {}, c1 = {}, c2 = {}, c3 = {};
        const float* aptr = Xg + nn * XG_LD + koff;
        const float* bb   = W1 + (size_t)(g * OG1 + obase0 + nn) * IG1 + koff;
        #pragma unroll 4
        for (int k0 = 0; k0 < IG1; k0 += 4) {
            v2f a  = *(const v2f*)(aptr + k0);
            v2f b0 = *(const v2f*)(bb + 0 * 16 * IG1 + k0);
            v2f b1v= *(const v2f*)(bb + 1 * 16 * IG1 + k0);
            v2f b2v= *(const v2f*)(bb + 2 * 16 * IG1 + k0);
            v2f b3 = *(const v2f*)(bb + 3 * 16 * IG1 + k0);
            c0 = __builtin_amdgcn_wmma_f32_16x16x4_f32(false, a, false, b0,  (short)0, c0, false, false);
            c1 = __builtin_amdgcn_wmma_f32_16x16x4_f32(false, a, false, b1v, (short)0, c1, false, false);
            c2 = __builtin_amdgcn_wmma_f32_16x16x4_f32(false, a, false, b2v, (short)0, c2, false, false);
            c3 = __builtin_amdgcn_wmma_f32_16x16x4_f32(false, a, false, b3,  (short)0, c3, false, false);
        }
        v8f cc[4] = {c0, c1, c2, c3};
        #pragma unroll
        for (int u = 0; u < 4; ++u) {
            const int obase = obase0 + u * 16;
            const float bias  = b1[g * OG1 + obase + nn];
            const int   cbase = (obase + nn) * NGRP + g;   // interleave: o*G + g
            #pragma unroll
            for (int r = 0; r < 8; ++r) {
                int m = r + 8 * hh;
                H[m * H_LD + cbase] = hardswish_f(cc[u][r] + bias);
            }
        }
    }
    __syncthreads();  // H complete & visible

    // ================= Layer 2: H[16,4608] x W2 -> out[16,1152] (interleaved)
    // Phase A: wave w owns all 8 output tiles of group g2 = w (A reused 8x).
    {
        const int g2 = wave;                     // 0..7
        v8f c[8];
        #pragma unroll
        for (int u = 0; u < 8; ++u) c[u] = (v8f){};
        const float* aptr = H  + nn * H_LD + g2 * IG2 + koff;
        const float* bb   = W2 + (size_t)(g2 * OG2 + nn) * IG2 + koff;
        #pragma unroll 2
        for (int k0 = 0; k0 < IG2; k0 += 4) {
            v2f a = *(const v2f*)(aptr + k0);
            #pragma unroll
            for (int u = 0; u < 8; ++u) {
                v2f b = *(const v2f*)(bb + (size_t)u * 16 * IG2 + k0);
                c[u] = __builtin_amdgcn_wmma_f32_16x16x4_f32(false, a, false, b, (short)0, c[u], false, false);
            }
        }
        #pragma unroll
        for (int u = 0; u < 8; ++u) {
            const int obase  = u * 16;
            const float bias = b2[g2 * OG2 + obase + nn];
            const int  cbase = (obase + nn) * NGRP + g2;   // interleave: o*G + g2
            #pragma unroll
            for (int r = 0; r < 8; ++r) {
                int m = r + 8 * hh;
                out[(token_base + m) * C_IN + cbase] = c[u][r] + bias;
            }
        }
    }
    // Phase B: group g2 = 8, tile w per wave (keeps all waves balanced: 9 tiles each).
    {
        const int g2    = 8;
        const int obase = wave * 16;
        v8f c = {};
        const float* aptr = H  + nn * H_LD + g2 * IG2 + koff;
        const float* bptr = W2 + (size_t)(g2 * OG2 + obase + nn) * IG2 + koff;
        #pragma unroll 8
        for (int k0 = 0; k0 < IG2; k0 += 4) {
            v2f a = *(const v2f*)(aptr + k0);
            v2f b = *(const v2f*)(bptr + k0);
            c = __builtin_amdgcn_wmma_f32_16x16x4_f32(false, a, false, b, (short)0, c, false, false);
        }
        const float bias  = b2[g2 * OG2 + obase + nn];
        const int   cbase = (obase + nn) * NGRP + g2;
        #pragma unroll
        for (int r = 0; r < 8; ++r) {
            int m = r + 8 * hh;
            out[(token_base + m) * C_IN + cbase] = c[r] + bias;
        }
    }
}

extern "C" void kernel_launch(void* const* d_in, const int* in_sizes, int n_in,
                              void* d_out, int out_size, void* d_ws, size_t ws_size,
                              hipStream_t stream) {
    (void)n_in; (void)out_size; (void)d_ws; (void)ws_size;
    const float* x  = (const float*)d_in[0];
    const float* W1 = (const float*)d_in[1];
    const float* b1 = (const float*)d_in[2];
    const float* W2 = (const float*)d_in[3];
    const float* b2 = (const float*)d_in[4];
    float* out = (float*)d_out;

    const int tokens  = in_sizes[0] / C_IN;   // 16384
    const int n_tiles = tokens / TM;          // 1024

    const size_t shmem = SMEM_FLOATS * sizeof(float);  // ~303 KB (<= 320 KB WGP LDS)
    (void)hipFuncSetAttribute((const void*)fused_local_mlp,
                              hipFuncAttributeMaxDynamicSharedMemorySize,
                              (int)shmem);

    fused_local_mlp<<<n_tiles, 256, shmem, stream>>>(x, W1, b1, W2, b2, out);
}